// DeepLinearAttention_35364760715518
// MI455X (gfx1250) — compile-verified
//
#include <hip/hip_runtime.h>
#include <hip/hip_bf16.h>
#include <math.h>

// Problem constants
#define Bn 8
#define Tn 256
#define INn 128
#define Hn 128
#define OUTn 128
#define Ln 2
#define STSTRIDE 33025      // L*(H*H+H)+1
#define LAYERSTATE 16512    // H*H+H
#define YSZ (Tn*Bn*OUTn)    // 262144

typedef __attribute__((ext_vector_type(2))) float v2f;
typedef __attribute__((ext_vector_type(8))) float v8f;

__device__ __forceinline__ v8f wmma4(v2f a, v2f b, v8f c) {
  // D = A(16x4 f32) * B(4x16 f32) + C(16x16 f32)
  return __builtin_amdgcn_wmma_f32_16x16x4_f32(false, a, false, b, (short)0, c, false, false);
}

// C(16x16) += A(16xK, row-major) * B^T where Bt is (16xK row-major) holding B's columns as rows.
__device__ __forceinline__ void tile_abt(const float* A, int lda, const float* Bt, int ldb,
                                         int K, v8f& acc) {
  const int lane = threadIdx.x & 31;
  const int half = lane >> 4;     // 0: K pair {0,1}; 1: K pair {2,3}
  const int idx  = lane & 15;     // row (A) / col (B)
  const float* ap = A  + (size_t)idx * lda + 2 * half;
  const float* bp = Bt + (size_t)idx * ldb + 2 * half;
  for (int k = 0; k < K; k += 4) {
    v2f a, b;
    a.x = ap[k]; a.y = ap[k + 1];
    b.x = bp[k]; b.y = bp[k + 1];
    acc = wmma4(a, b, acc);
  }
}

// C(16x16) += A(16xK, row-major) * B(KxN row-major), tile col base already applied to Bm.
__device__ __forceinline__ void tile_ab(const float* A, int lda, const float* Bm, int ldb,
                                        int K, v8f& acc) {
  const int lane = threadIdx.x & 31;
  const int half = lane >> 4;
  const int idx  = lane & 15;
  const float* ap = A + (size_t)idx * lda + 2 * half;
  for (int k = 0; k < K; k += 4) {
    v2f a, b;
    a.x = ap[k]; a.y = ap[k + 1];
    b.x = Bm[(size_t)(k + 2 * half)     * ldb + idx];
    b.y = Bm[(size_t)(k + 2 * half + 1) * ldb + idx];
    acc = wmma4(a, b, acc);
  }
}

// C(16x16) += A^T * B, logical A[m][k] = Am[k*lda + m] (Am has tile row base applied as col offset),
// B(KxN row-major) with tile col base applied.
__device__ __forceinline__ void tile_atb(const float* Am, int lda, const float* Bm, int ldb,
                                         int K, v8f& acc) {
  const int lane = threadIdx.x & 31;
  const int half = lane >> 4;
  const int idx  = lane & 15;
  for (int k = 0; k < K; k += 4) {
    v2f a, b;
    a.x = Am[(size_t)(k + 2 * half)     * lda + idx];
    a.y = Am[(size_t)(k + 2 * half + 1) * lda + idx];
    b.x = Bm[(size_t)(k + 2 * half)     * ldb + idx];
    b.y = Bm[(size_t)(k + 2 * half + 1) * ldb + idx];
    acc = wmma4(a, b, acc);
  }
}

// ---------------- rotary embedding + ts update ----------------
__global__ void dla_rotate_kernel(const float* __restrict__ z, const float* __restrict__ st,
                                  float* __restrict__ xcur, float* __restrict__ outst) {
  const int bt = blockIdx.x;          // B*T
  const int b = bt >> 8, t = bt & 255;
  const int j = threadIdx.x;          // 0..63
  const float ts = st[(size_t)b * STSTRIDE + STSTRIDE - 1];
  const float pos = ts + (float)t;
  // freq_j = exp(-(2j) * ln(10000) / 128)
  const float freq = expf((float)(2 * j) * (-9.210340371976184f / 128.0f));
  const float ang = pos * freq;
  const float s = sinf(ang), c = cosf(ang);
  const float* zp = z + (size_t)t * Bn * INn + (size_t)b * INn;
  const float x1 = zp[j], x2 = zp[j + 64];
  float* xp = xcur + ((size_t)b * Tn + t) * INn;
  xp[j]      = x1 * c - x2 * s;
  xp[j + 64] = x2 * c + x1 * s;
  if (t == 0 && j == 0)
    outst[(size_t)b * STSTRIDE + STSTRIDE - 1] = ts + (float)Tn;
}

// ---------------- layernorm (one wave per row) ----------------
__global__ void dla_ln_kernel(const float* __restrict__ x, const float* __restrict__ w,
                              const float* __restrict__ bias, float* __restrict__ xn) {
  const int wave = threadIdx.x >> 5, lane = threadIdx.x & 31;
  const int row = blockIdx.x * 8 + wave;    // 0..B*T-1
  const float* xr = x + (size_t)row * INn;
  float s = 0.f, ss = 0.f;
  for (int k = lane; k < INn; k += 32) { float v = xr[k]; s += v; ss += v * v; }
  for (int off = 16; off; off >>= 1) { s += __shfl_xor(s, off); ss += __shfl_xor(ss, off); }
  const float m = s * (1.0f / INn);
  const float var = ss * (1.0f / INn) - m * m;
  const float inv = rsqrtf(var + 1e-5f);
  float* outr = xn + (size_t)row * INn;
  for (int k = lane; k < INn; k += 32)
    outr[k] = (xr[k] - m) * inv * w[k] + bias[k];
}

// ---------------- K/Q/V projections (elu+1 on K,Q) ----------------
__global__ void dla_proj_kernel(const float* __restrict__ xn,
                                const float* __restrict__ Wk, const float* __restrict__ Wq,
                                const float* __restrict__ Wv,
                                float* __restrict__ Kb, float* __restrict__ Qb,
                                float* __restrict__ Vb) {
  const int g = blockIdx.x * 8 + (threadIdx.x >> 5);   // 3*B*128 tiles
  const int which = g >> 10;           // /1024
  const int rem = g & 1023;
  const int b = rem >> 7;
  const int tile = rem & 127;
  const int tm = tile >> 3, tn = tile & 7;
  const float* W = (which == 0) ? Wk : (which == 1) ? Wq : Wv;
  const float* A  = xn + (size_t)b * Tn * INn + (size_t)tm * 16 * INn;
  const float* Bt = W + (size_t)tn * 16 * INn;
  v8f acc = {0.f,0.f,0.f,0.f,0.f,0.f,0.f,0.f};
  tile_abt(A, INn, Bt, INn, INn, acc);
  float* Out = ((which == 0) ? Kb : (which == 1) ? Qb : Vb)
             + (size_t)b * Tn * Hn + (size_t)tm * 16 * Hn + tn * 16;
  const int lane = threadIdx.x & 31, half = lane >> 4, idx = lane & 15;
#pragma unroll
  for (int r = 0; r < 8; r++) {
    float v = acc[r];
    if (which != 2) v = ((v > 0.f) ? v : expm1f(v)) + 1.0f;   // elu(v)+1
    Out[(size_t)(r + 8 * half) * Hn + idx] = v;
  }
}

// ---------------- causal scores A = tril(Q K^T) ----------------
__global__ void dla_score_kernel(const float* __restrict__ Qb, const float* __restrict__ Kb,
                                 float* __restrict__ Ab) {
  const int g = blockIdx.x * 8 + (threadIdx.x >> 5);   // B*256 tiles
  const int b = g >> 8;
  const int tile = g & 255;
  const int tm = tile >> 4, tn = tile & 15;
  const int lane = threadIdx.x & 31, half = lane >> 4, idx = lane & 15;
  float* Out = Ab + (size_t)b * Tn * Tn + (size_t)tm * 16 * Tn + tn * 16;
  if (tn > tm) {       // strictly above diagonal -> zeros
#pragma unroll
    for (int r = 0; r < 8; r++) Out[(size_t)(r + 8 * half) * Tn + idx] = 0.f;
    return;
  }
  v8f acc = {0.f,0.f,0.f,0.f,0.f,0.f,0.f,0.f};
  tile_abt(Qb + (size_t)b * Tn * Hn + (size_t)tm * 16 * Hn, Hn,
           Kb + (size_t)b * Tn * Hn + (size_t)tn * 16 * Hn, Hn, Hn, acc);
#pragma unroll
  for (int r = 0; r < 8; r++) {
    float v = acc[r];
    if (tn == tm && idx > (r + 8 * half)) v = 0.f;   // mask col > row on diagonal tile
    Out[(size_t)(r + 8 * half) * Tn + idx] = v;
  }
}

// ---------------- den[t] = rowsum(A[t,:]) + Q[t].Z0 + 1e-5 ----------------
__global__ void dla_den_kernel(const float* __restrict__ Ab, const float* __restrict__ Qb,
                               const float* __restrict__ st, int layerOff,
                               float* __restrict__ den) {
  const int wave = threadIdx.x >> 5, lane = threadIdx.x & 31;
  const int row = blockIdx.x * 8 + wave;     // 0..B*T-1
  const int b = row >> 8;
  const float* Ar = Ab + (size_t)b * Tn * Tn + (size_t)(row & 255) * Tn;
  const float* Qr = Qb + (size_t)row * Hn;
  const float* Z0 = st + (size_t)b * STSTRIDE + layerOff + Hn * Hn;
  float s = 0.f;
  for (int k = lane; k < Tn; k += 32) s += Ar[k];
  for (int k = lane; k < Hn; k += 32) s += Qr[k] * Z0[k];
  for (int off = 16; off; off >>= 1) s += __shfl_xor(s, off);
  if (lane == 0) den[row] = s + 1e-5f;
}

// ---------------- attn out = (A@V + Q@S0) / den ----------------
__global__ void dla_attnout_kernel(const float* __restrict__ Ab, const float* __restrict__ Vb,
                                   const float* __restrict__ Qb, const float* __restrict__ st,
                                   int layerOff, const float* __restrict__ den,
                                   float* __restrict__ Ao) {
  const int g = blockIdx.x * 8 + (threadIdx.x >> 5);   // B*128 tiles
  const int b = g >> 7;
  const int tile = g & 127;
  const int tm = tile >> 3, tn = tile & 7;
  v8f acc = {0.f,0.f,0.f,0.f,0.f,0.f,0.f,0.f};
  // A @ V  (K = T)
  tile_ab(Ab + (size_t)b * Tn * Tn + (size_t)tm * 16 * Tn, Tn,
          Vb + (size_t)b * Tn * Hn + tn * 16, Hn, Tn, acc);
  // Q @ S0 (K = H)
  const float* S0 = st + (size_t)b * STSTRIDE + layerOff;
  tile_ab(Qb + (size_t)b * Tn * Hn + (size_t)tm * 16 * Hn, Hn,
          S0 + tn * 16, Hn, Hn, acc);
  const int lane = threadIdx.x & 31, half = lane >> 4, idx = lane & 15;
  float* Out = Ao + (size_t)b * Tn * Hn + (size_t)tm * 16 * Hn + tn * 16;
#pragma unroll
  for (int r = 0; r < 8; r++) {
    const int row = tm * 16 + r + 8 * half;
    Out[(size_t)(r + 8 * half) * Hn + idx] = acc[r] / den[b * Tn + row];
  }
}

// ---------------- S' = S0 + K^T V ----------------
__global__ void dla_states_kernel(const float* __restrict__ Kb, const float* __restrict__ Vb,
                                  const float* __restrict__ st, int layerOff,
                                  float* __restrict__ outst) {
  const int g = blockIdx.x * 8 + (threadIdx.x >> 5);   // B*64 tiles
  const int b = g >> 6;
  const int tile = g & 63;
  const int tm = tile >> 3, tn = tile & 7;
  v8f acc = {0.f,0.f,0.f,0.f,0.f,0.f,0.f,0.f};
  tile_atb(Kb + (size_t)b * Tn * Hn + tm * 16, Hn,
           Vb + (size_t)b * Tn * Hn + tn * 16, Hn, Tn, acc);
  const float* S0 = st + (size_t)b * STSTRIDE + layerOff;
  float* Out = outst + (size_t)b * STSTRIDE + layerOff;
  const int lane = threadIdx.x & 31, half = lane >> 4, idx = lane & 15;
#pragma unroll
  for (int r = 0; r < 8; r++) {
    const int i = tm * 16 + r + 8 * half;
    const int l = tn * 16 + idx;
    Out[(size_t)i * Hn + l] = acc[r] + S0[(size_t)i * Hn + l];
  }
}

// ---------------- Z' = Z0 + sum_t K[t] ----------------
__global__ void dla_statez_kernel(const float* __restrict__ Kb, const float* __restrict__ st,
                                  int layerOff, float* __restrict__ outst) {
  const int b = blockIdx.x, i = threadIdx.x;
  const float* Kp = Kb + (size_t)b * Tn * Hn + i;
  float s = 0.f;
  for (int t = 0; t < Tn; t++) s += Kp[(size_t)t * Hn];
  const size_t o = (size_t)b * STSTRIDE + layerOff + Hn * Hn + i;
  outst[o] = st[o] + s;
}

// ---------------- h1 = relu(attn @ w1^T + b1) ----------------
__global__ void dla_ff1_kernel(const float* __restrict__ Ao, const float* __restrict__ w1,
                               const float* __restrict__ b1, float* __restrict__ h1) {
  const int g = blockIdx.x * 8 + (threadIdx.x >> 5);   // B*128 tiles
  const int b = g >> 7;
  const int tile = g & 127;
  const int tm = tile >> 3, tn = tile & 7;
  v8f acc = {0.f,0.f,0.f,0.f,0.f,0.f,0.f,0.f};
  tile_abt(Ao + (size_t)b * Tn * Hn + (size_t)tm * 16 * Hn, Hn,
           w1 + (size_t)tn * 16 * Hn, Hn, Hn, acc);
  const int lane = threadIdx.x & 31, half = lane >> 4, idx = lane & 15;
  float* Out = h1 + (size_t)b * Tn * Hn + (size_t)tm * 16 * Hn + tn * 16;
#pragma unroll
  for (int r = 0; r < 8; r++) {
    float v = acc[r] + b1[tn * 16 + idx];
    Out[(size_t)(r + 8 * half) * Hn + idx] = v > 0.f ? v : 0.f;
  }
}

// ---------------- xnew = relu(h1 @ w2^T + b2) + xn @ scw^T + scb ----------------
__global__ void dla_ff2_kernel(const float* __restrict__ h1, const float* __restrict__ w2,
                               const float* __restrict__ b2, const float* __restrict__ xn,
                               const float* __restrict__ scw, const float* __restrict__ scb,
                               float* __restrict__ xcur) {
  const int g = blockIdx.x * 8 + (threadIdx.x >> 5);   // B*128 tiles
  const int b = g >> 7;
  const int tile = g & 127;
  const int tm = tile >> 3, tn = tile & 7;
  v8f acc1 = {0.f,0.f,0.f,0.f,0.f,0.f,0.f,0.f};
  v8f acc2 = {0.f,0.f,0.f,0.f,0.f,0.f,0.f,0.f};
  tile_abt(h1 + (size_t)b * Tn * Hn + (size_t)tm * 16 * Hn, Hn,
           w2 + (size_t)tn * 16 * Hn, Hn, Hn, acc1);
  tile_abt(xn + (size_t)b * Tn * INn + (size_t)tm * 16 * INn, INn,
           scw + (size_t)tn * 16 * INn, INn, INn, acc2);
  const int lane = threadIdx.x & 31, half = lane >> 4, idx = lane & 15;
  float* Out = xcur + (size_t)b * Tn * Hn + (size_t)tm * 16 * Hn + tn * 16;
#pragma unroll
  for (int r = 0; r < 8; r++) {
    float v = acc1[r] + b2[tn * 16 + idx];
    v = v > 0.f ? v : 0.f;
    Out[(size_t)(r + 8 * half) * Hn + idx] = v + acc2[r] + scb[tn * 16 + idx];
  }
}

// ---------------- y = x @ unmap_w^T + unmap_b, stored (T,B,OUT) ----------------
__global__ void dla_unmap_kernel(const float* __restrict__ x, const float* __restrict__ uw,
                                 const float* __restrict__ ub, float* __restrict__ y) {
  const int g = blockIdx.x * 8 + (threadIdx.x >> 5);   // B*128 tiles
  const int b = g >> 7;
  const int tile = g & 127;
  const int tm = tile >> 3, tn = tile & 7;
  v8f acc = {0.f,0.f,0.f,0.f,0.f,0.f,0.f,0.f};
  tile_abt(x + (size_t)b * Tn * Hn + (size_t)tm * 16 * Hn, Hn,
           uw + (size_t)tn * 16 * Hn, Hn, Hn, acc);
  const int lane = threadIdx.x & 31, half = lane >> 4, idx = lane & 15;
#pragma unroll
  for (int r = 0; r < 8; r++) {
    const int t = tm * 16 + r + 8 * half;
    const int o = tn * 16 + idx;
    y[(size_t)t * Bn * OUTn + (size_t)b * OUTn + o] = acc[r] + ub[o];
  }
}

extern "C" void kernel_launch(void* const* d_in, const int* in_sizes, int n_in,
                              void* d_out, int out_size, void* d_ws, size_t ws_size,
                              hipStream_t stream) {
  (void)in_sizes; (void)n_in; (void)out_size; (void)ws_size;
  const float* z   = (const float*)d_in[0];
  const float* st  = (const float*)d_in[1];
  const float* Wk  = (const float*)d_in[2];
  const float* Wq  = (const float*)d_in[3];
  const float* Wv  = (const float*)d_in[4];
  const float* lnw = (const float*)d_in[5];
  const float* lnb = (const float*)d_in[6];
  const float* w1  = (const float*)d_in[7];
  const float* b1  = (const float*)d_in[8];
  const float* w2  = (const float*)d_in[9];
  const float* b2  = (const float*)d_in[10];
  const float* scw = (const float*)d_in[11];
  const float* scb = (const float*)d_in[12];
  const float* uw  = (const float*)d_in[13];
  const float* ub  = (const float*)d_in[14];
  float* out = (float*)d_out;
  float* ws  = (float*)d_ws;

  float* XCUR = ws;                 // B*T*H
  float* XN   = ws + 262144;        // B*T*H
  float* KB   = ws + 524288;        // B*T*H
  float* QB   = ws + 786432;        // B*T*H (reused as h1 in FF)
  float* VB   = ws + 1048576;       // B*T*H
  float* AB   = ws + 1310720;       // B*T*T
  float* AO   = ws + 1835008;       // B*T*H
  float* DEN  = ws + 2097152;       // B*T
  float* OST  = out + YSZ;          // state output region

  dla_rotate_kernel<<<Bn * Tn, 64, 0, stream>>>(z, st, XCUR, OST);
  for (int i = 0; i < Ln; i++) {
    const int lo = i * LAYERSTATE;
    dla_ln_kernel<<<256, 256, 0, stream>>>(XCUR, lnw + i * INn, lnb + i * INn, XN);
    dla_proj_kernel<<<384, 256, 0, stream>>>(XN, Wk + (size_t)i * Hn * INn,
                                             Wq + (size_t)i * Hn * INn,
                                             Wv + (size_t)i * Hn * INn, KB, QB, VB);
    dla_score_kernel<<<256, 256, 0, stream>>>(QB, KB, AB);
    dla_den_kernel<<<256, 256, 0, stream>>>(AB, QB, st, lo, DEN);
    dla_attnout_kernel<<<128, 256, 0, stream>>>(AB, VB, QB, st, lo, DEN, AO);
    dla_states_kernel<<<64, 256, 0, stream>>>(KB, VB, st, lo, OST);
    dla_statez_kernel<<<8, 128, 0, stream>>>(KB, st, lo, OST);
    dla_ff1_kernel<<<128, 256, 0, stream>>>(AO, w1 + (size_t)i * Hn * Hn, b1 + i * Hn, QB);
    dla_ff2_kernel<<<128, 256, 0, stream>>>(QB, w2 + (size_t)i * Hn * Hn, b2 + i * Hn,
                                            XN, scw + (size_t)i * Hn * INn, scb + i * Hn, XCUR);
  }
  dla_unmap_kernel<<<128, 256, 0, stream>>>(XCUR, uw, ub, out);
}